// RITSI_55490977464473
// MI455X (gfx1250) — compile-verified
//
#include <hip/hip_runtime.h>

// ---------------------------------------------------------------------------
// RITS / GRU-D recurrence for MI455X (gfx1250), wave32, fp32 WMMA 16x16x4.
// One workgroup (4 waves) owns a 16-row batch tile and walks the sequence.
// Weights resident in LDS; x/m/d tiles double-buffered and prefetched one
// step ahead (registers -> LDS during the LSTM phase) to hide HBM latency.
// Gates GEMM is split: the mask-half runs concurrently with the decay phase.
// (Register-resident weight fragments were tried and reverted: they pushed
// allocation past 256 VGPRs into VGPR-MSB mode, tripling code size.)
// ---------------------------------------------------------------------------

typedef float v2f __attribute__((ext_vector_type(2)));
typedef float v8f __attribute__((ext_vector_type(8)));
typedef float f4  __attribute__((ext_vector_type(4)));

#define WMMA_F32(a, b, c) \
    __builtin_amdgcn_wmma_f32_16x16x4_f32(false, (a), false, (b), (short)0, (c), false, false)

constexpr int Bsz = 256, S = 1024, D = 64, H = 32;
constexpr int BT  = 16;    // batch rows per workgroup (WMMA M)
constexpr int NT  = 128;   // threads per block = 4 wave32

__device__ __forceinline__ float fast_sigmoid(float x) {
    return 1.0f / (1.0f + __expf(-x));
}
__device__ __forceinline__ float fast_tanh(float x) {
    return 1.0f - 2.0f / (__expf(2.0f * x) + 1.0f);   // saturates correctly
}
__device__ __forceinline__ f4 ntload4(const float* p) {
    return __builtin_nontemporal_load((const f4*)p);
}
__device__ __forceinline__ void ntstore1(float* p, float v) {
    __builtin_nontemporal_store(v, p);
}

__global__ __launch_bounds__(NT) void rits_scan_kernel(
    const float* __restrict__ values, const float* __restrict__ masks,
    const float* __restrict__ deltas,
    const float* __restrict__ Wd,  const float* __restrict__ bd,
    const float* __restrict__ Wr,  const float* __restrict__ br,
    const float* __restrict__ Wih, const float* __restrict__ bih,
    const float* __restrict__ Whh, const float* __restrict__ bhh,
    float* __restrict__ out)
{
    // ---- LDS: weights (resident) + double-buffered staging (~136 KB) ----
    __shared__ __align__(16) float sWih[128 * 128];   // 64 KB
    __shared__ __align__(16) float sWhh[128 * 32];    // 16 KB
    __shared__ __align__(16) float sWd [32 * 64];     //  8 KB
    __shared__ __align__(16) float sWr [64 * 32];     //  8 KB
    __shared__ __align__(16) float sX [2][BT * 64];   // values tiles (db)
    __shared__ __align__(16) float sM [2][BT * 64];   // mask tiles   (db)
    __shared__ __align__(16) float sD [2][BT * 64];   // delta tiles  (db)
    __shared__ __align__(16) float sXc[BT * 64];      // imputed x_c (A-matrix)
    __shared__ __align__(16) float sGat[BT * 128];    // gates staging
    __shared__ __align__(16) float sH [BT * 32];      // hidden state
    __shared__ __align__(16) float sC [BT * 32];      // cell state

    const int tid  = threadIdx.x;
    const int wave = tid >> 5;
    const int lane = tid & 31;
    const int hi   = lane >> 4;    // half-wave: selects K pair / M half
    const int ln   = lane & 15;    // A row index == B column index
    const int bb   = blockIdx.x * BT;

    // ---- one-time: weights -> LDS (coalesced float4), state = 0 ----
    for (int i = tid * 4; i < 128 * 128; i += NT * 4)
        *(f4*)(sWih + i) = *(const f4*)(Wih + i);
    for (int i = tid * 4; i < 128 * 32;  i += NT * 4)
        *(f4*)(sWhh + i) = *(const f4*)(Whh + i);
    for (int i = tid * 4; i < 32 * 64;   i += NT * 4)
        *(f4*)(sWd  + i) = *(const f4*)(Wd  + i);
    for (int i = tid * 4; i < 64 * 32;   i += NT * 4)
        *(f4*)(sWr  + i) = *(const f4*)(Wr  + i);
    for (int i = tid; i < BT * 32; i += NT) { sH[i] = 0.0f; sC[i] = 0.0f; }

    // ---- hoisted per-wave constants ----
    const float bias_dec = (wave < 2) ? bd[wave * 16 + ln] : 0.0f;  // H=32
    const float bias_reg = br[wave * 16 + ln];                      // D=64
    const int   n0a = (2 * wave) * 16, n0b = n0a + 16;              // gate tiles
    const float bias_ga = bih[n0a + ln] + bhh[n0a + ln];
    const float bias_gb = bih[n0b + ln] + bhh[n0b + ln];

    // ---- staging geometry: each thread owns 8 consecutive floats ----
    const int    prow = tid >> 3;
    const int    pcol = (tid & 7) * 8;
    const size_t grow = (size_t)(bb + prow) * (size_t)S * (size_t)D + pcol;

    // ---- prologue: stage t = 0 into buffer 0 ----
    {
        const size_t g = grow;    // t = 0
        *(f4*)(&sX[0][prow * 64 + pcol])     = ntload4(values + g);
        *(f4*)(&sX[0][prow * 64 + pcol + 4]) = ntload4(values + g + 4);
        *(f4*)(&sM[0][prow * 64 + pcol])     = ntload4(masks + g);
        *(f4*)(&sM[0][prow * 64 + pcol + 4]) = ntload4(masks + g + 4);
        *(f4*)(&sD[0][prow * 64 + pcol])     = ntload4(deltas + g);
        *(f4*)(&sD[0][prow * 64 + pcol + 4]) = ntload4(deltas + g + 4);
    }
    __syncthreads();

    for (int t = 0; t < S; ++t) {
        const int  cur = t & 1, nxt = cur ^ 1;
        const bool pf  = (t + 1 < S);

        // ---- issue prefetch for t+1 (committed to LDS in phase D) ----
        f4 px0{}, px1{}, pm0{}, pm1{}, pd0{}, pd1{};
        if (pf) {
            const size_t g = grow + (size_t)(t + 1) * (size_t)D;
            px0 = ntload4(values + g); px1 = ntload4(values + g + 4);
            pm0 = ntload4(masks  + g); pm1 = ntload4(masks  + g + 4);
            pd0 = ntload4(deltas + g); pd1 = ntload4(deltas + g + 4);
        }

        // ---- Phase A (waves 0,1): gamma = exp(-relu(d@Wd^T+bd)); h *= gamma
        if (wave < 2) {
            const int n0 = wave * 16;
            v8f acc;
            #pragma unroll
            for (int r = 0; r < 8; ++r) acc[r] = bias_dec;
            const float* wrow = sWd + (n0 + ln) * 64;
            #pragma unroll
            for (int kk = 0; kk < 16; ++kk) {               // K = 64
                const int k0 = 4 * kk + 2 * hi;
                v2f a = { sD[cur][ln * 64 + k0], sD[cur][ln * 64 + k0 + 1] };
                v2f b = { wrow[k0], wrow[k0 + 1] };
                acc = WMMA_F32(a, b, acc);
            }
            #pragma unroll
            for (int r = 0; r < 8; ++r) {
                const float gma = __expf(-fmaxf(acc[r], 0.0f));
                sH[(r + 8 * hi) * 32 + n0 + ln] *= gma;
            }
        }

        // ---- Phase C part 1 (all waves): mask-half of gate GEMM ----
        v8f acc0, acc1;
        #pragma unroll
        for (int r = 0; r < 8; ++r) { acc0[r] = bias_ga; acc1[r] = bias_gb; }
        {
            const float* wa = sWih + (n0a + ln) * 128 + 64;  // m columns 64..127
            const float* wb = sWih + (n0b + ln) * 128 + 64;
            #pragma unroll 8
            for (int kk = 0; kk < 16; ++kk) {               // K = 64 (m half)
                const int k0 = 4 * kk + 2 * hi;
                v2f a = { sM[cur][ln * 64 + k0], sM[cur][ln * 64 + k0 + 1] };
                v2f u = { wa[k0], wa[k0 + 1] };
                v2f v = { wb[k0], wb[k0 + 1] };
                acc0 = WMMA_F32(a, u, acc0);
                acc1 = WMMA_F32(a, v, acc1);
            }
        }
        __syncthreads();

        // ---- Phase B: x_h = h@Wr^T + br; x_c = m*x + (1-m)*x_h ----
        {
            const int n0 = wave * 16;                        // D cols 16w..16w+15
            v8f acc;
            #pragma unroll
            for (int r = 0; r < 8; ++r) acc[r] = bias_reg;
            const float* wrow = sWr + (n0 + ln) * 32;
            #pragma unroll
            for (int kk = 0; kk < 8; ++kk) {                // K = 32
                const int k0 = 4 * kk + 2 * hi;
                v2f a = { sH[ln * 32 + k0], sH[ln * 32 + k0 + 1] };
                v2f b = { wrow[k0], wrow[k0 + 1] };
                acc = WMMA_F32(a, b, acc);
            }
            #pragma unroll
            for (int r = 0; r < 8; ++r) {
                const int row = r + 8 * hi;
                const int col = n0 + ln;
                const float xv = sX[cur][row * 64 + col];
                const float mv = sM[cur][row * 64 + col];
                const float xc = mv * xv + (1.0f - mv) * acc[r];
                sXc[row * 64 + col] = xc;
                ntstore1(out + ((size_t)(bb + row) * S + t) * (size_t)D + col, xc);
            }
        }
        __syncthreads();

        // ---- Phase C part 2: x_c-half of gate GEMM + h@Whh^T ----
        {
            const float* wa = sWih + (n0a + ln) * 128;       // x_c columns 0..63
            const float* wb = sWih + (n0b + ln) * 128;
            #pragma unroll 8
            for (int kk = 0; kk < 16; ++kk) {               // K = 64 (x_c half)
                const int k0 = 4 * kk + 2 * hi;
                v2f a = { sXc[ln * 64 + k0], sXc[ln * 64 + k0 + 1] };
                v2f u = { wa[k0], wa[k0 + 1] };
                v2f v = { wb[k0], wb[k0 + 1] };
                acc0 = WMMA_F32(a, u, acc0);
                acc1 = WMMA_F32(a, v, acc1);
            }
            const float* ha = sWhh + (n0a + ln) * 32;
            const float* hb = sWhh + (n0b + ln) * 32;
            #pragma unroll
            for (int kk = 0; kk < 8; ++kk) {                // K = 32 (hidden)
                const int k0 = 4 * kk + 2 * hi;
                v2f a = { sH[ln * 32 + k0], sH[ln * 32 + k0 + 1] };
                v2f u = { ha[k0], ha[k0 + 1] };
                v2f v = { hb[k0], hb[k0 + 1] };
                acc0 = WMMA_F32(a, u, acc0);
                acc1 = WMMA_F32(a, v, acc1);
            }
            #pragma unroll
            for (int r = 0; r < 8; ++r) {
                const int row = r + 8 * hi;
                sGat[row * 128 + n0a + ln] = acc0[r];
                sGat[row * 128 + n0b + ln] = acc1[r];
            }
        }
        __syncthreads();

        // ---- Phase D: LSTM cell update + commit prefetched tiles ----
        #pragma unroll
        for (int q = 0; q < 4; ++q) {
            const int e   = tid + NT * q;                   // 0..511
            const int row = e >> 5;
            const int j   = e & 31;
            const float gi = sGat[row * 128 +      j];
            const float gf = sGat[row * 128 + 32 + j];
            const float gg = sGat[row * 128 + 64 + j];
            const float go = sGat[row * 128 + 96 + j];
            const float cprev = sC[row * 32 + j];
            const float cn = fast_sigmoid(gf) * cprev + fast_sigmoid(gi) * fast_tanh(gg);
            sC[row * 32 + j] = cn;
            sH[row * 32 + j] = fast_sigmoid(go) * fast_tanh(cn);
        }
        if (pf) {
            *(f4*)(&sX[nxt][prow * 64 + pcol])     = px0;
            *(f4*)(&sX[nxt][prow * 64 + pcol + 4]) = px1;
            *(f4*)(&sM[nxt][prow * 64 + pcol])     = pm0;
            *(f4*)(&sM[nxt][prow * 64 + pcol + 4]) = pm1;
            *(f4*)(&sD[nxt][prow * 64 + pcol])     = pd0;
            *(f4*)(&sD[nxt][prow * 64 + pcol + 4]) = pd1;
        }
        __syncthreads();
    }
}

extern "C" void kernel_launch(void* const* d_in, const int* in_sizes, int n_in,
                              void* d_out, int out_size, void* d_ws, size_t ws_size,
                              hipStream_t stream) {
    const float* values = (const float*)d_in[0];
    const float* masks  = (const float*)d_in[1];
    const float* deltas = (const float*)d_in[2];
    const float* Wd     = (const float*)d_in[3];
    const float* bd     = (const float*)d_in[4];
    const float* Wr     = (const float*)d_in[5];
    const float* br     = (const float*)d_in[6];
    const float* Wih    = (const float*)d_in[7];
    const float* bih    = (const float*)d_in[8];
    const float* Whh    = (const float*)d_in[9];
    const float* bhh    = (const float*)d_in[10];
    float* out = (float*)d_out;

    rits_scan_kernel<<<dim3(Bsz / BT), dim3(NT), 0, stream>>>(
        values, masks, deltas, Wd, bd, Wr, br, Wih, bih, Whh, bhh, out);
}